// PolynomialModel_49933289783886
// MI455X (gfx1250) — compile-verified
//
#include <hip/hip_runtime.h>
#include <hip/hip_bf16.h>

typedef __attribute__((ext_vector_type(2))) float v2f;
typedef __attribute__((ext_vector_type(8))) float v8f;

// y(x) = sum_{i=0}^{1000} w[i] x^i, i = 16j + k (w zero-padded to 1024).
// Per 16-element tile, compute C = W^T x P^T via V_WMMA_F32_16X16X4_F32:
//   A_t[M][k] = w[16*(16t+M) + k]   (M = block row, 4 tiles t, K split in 4)
//   B  [k][e] = x_e^k               (N = elements -> columns of C)
// C layout then gives lane l rows j = 16t + 8*hi + r of ITS OWN element
// (e = l&15), so the final y_e = sum_j C[j][e] z^j  (z = x^16) is a pure
// per-lane double Horner + one cross-half shuffle.
__global__ void poly_wmma_kernel(const float* __restrict__ x,
                                 const float* __restrict__ w,
                                 float* __restrict__ out,
                                 int nTiles, int nElems, int nW) {
    const int lane          = threadIdx.x & 31;
    const int waveInBlock   = threadIdx.x >> 5;
    const int wavesPerBlock = blockDim.x >> 5;
    const int gwave         = blockIdx.x * wavesPerBlock + waveInBlock;
    const int nWaves        = gridDim.x * wavesPerBlock;

    const int  n     = lane & 15;         // element slot (B column) / A row
    const int  khalf = (lane >> 4) << 1;  // 0 for lanes 0-15, 2 for lanes 16-31
    const bool hiH   = (lane >= 16);

    // ---- Preload A fragments (W^T), reused for every tile this wave does.
    // A tile t, K-chunk c: lane holds {w[256t+16n+4c+khalf], w[...+1]}
    v2f Af[4][4];
#pragma unroll
    for (int t = 0; t < 4; ++t) {
#pragma unroll
        for (int c = 0; c < 4; ++c) {
            const int idx0 = 256 * t + 16 * n + 4 * c + khalf;
            const float a0 = (idx0     < nW) ? w[idx0]     : 0.0f;
            const float a1 = (idx0 + 1 < nW) ? w[idx0 + 1] : 0.0f;
            v2f a = {a0, a1};
            Af[t][c] = a;
        }
    }

    for (int tile = gwave; tile < nTiles; tile += nWaves) {
        // Lanes l and l+16 both carry element e = (l&15) of this tile.
        const int   ei = tile * 16 + n;
        const float xe = (ei < nElems) ? x[ei] : 0.0f;
        const float x2 = xe * xe;
        const float s  = x2 * x2;                 // x^4: B-chunk advance

        // B chunk c=0: lanes 0-15 hold {x^0, x^1}; lanes 16-31 hold {x^2, x^3}
        float b0 = hiH ? x2 : 1.0f;
        float b1 = b0 * xe;

        v8f acc0 = {}, acc1 = {}, acc2 = {}, acc3 = {};
#pragma unroll
        for (int c = 0; c < 4; ++c) {
            v2f B = {b0, b1};
            acc0 = __builtin_amdgcn_wmma_f32_16x16x4_f32(
                false, Af[0][c], false, B, (short)0, acc0, false, false);
            acc1 = __builtin_amdgcn_wmma_f32_16x16x4_f32(
                false, Af[1][c], false, B, (short)0, acc1, false, false);
            acc2 = __builtin_amdgcn_wmma_f32_16x16x4_f32(
                false, Af[2][c], false, B, (short)0, acc2, false, false);
            acc3 = __builtin_amdgcn_wmma_f32_16x16x4_f32(
                false, Af[3][c], false, B, (short)0, acc3, false, false);
            b0 *= s;                               // advance to next K-chunk
            b1 *= s;
        }

        // z = x^16, z8 = x^128, Z = z^16 = x^256 (all for this lane's element)
        const float x8 = s * s;
        const float z  = x8 * x8;
        const float zz2 = z * z;
        const float zz4 = zz2 * zz2;
        const float z8  = zz4 * zz4;
        const float Zl  = z8 * z8;

        // acc_t[r] = C[16t + 8*hiH + r][e].  Double Horner:
        //   y_half = sum_r z^r * (sum_t Z^t * acc_t[r]),  hi half * z^8
        float part = 0.0f;
#pragma unroll
        for (int r = 7; r >= 0; --r) {
            float Ar = fmaf(acc3[r], Zl, acc2[r]);
            Ar       = fmaf(Ar,      Zl, acc1[r]);
            Ar       = fmaf(Ar,      Zl, acc0[r]);
            part     = fmaf(part, z, Ar);
        }
        float val = hiH ? part * z8 : part;
        val += __shfl_xor(val, 16, 32);           // merge the two halves

        const int oi = tile * 16 + lane;
        if (lane < 16 && oi < nElems) out[oi] = val;
    }
}

extern "C" void kernel_launch(void* const* d_in, const int* in_sizes, int n_in,
                              void* d_out, int out_size, void* d_ws, size_t ws_size,
                              hipStream_t stream) {
    const float* x  = (const float*)d_in[0];   // 512*1024 f32
    const float* w  = (const float*)d_in[1];   // 1001 f32
    float*       o  = (float*)d_out;

    const int nElems = in_sizes[0];
    const int nW     = in_sizes[1];
    const int nTiles = (nElems + 15) / 16;     // 32768 tiles of 16 elements

    const int threads       = 256;             // 8 wave32 per block
    const int wavesPerBlock = threads / 32;
    const int tilesPerWave  = 8;               // amortize A-fragment preload
    int blocks = (nTiles + wavesPerBlock * tilesPerWave - 1) /
                 (wavesPerBlock * tilesPerWave);
    if (blocks < 1) blocks = 1;

    poly_wmma_kernel<<<blocks, threads, 0, stream>>>(x, w, o, nTiles, nElems, nW);
}